// CCG_46273977647541
// MI455X (gfx1250) — compile-verified
//
#include <hip/hip_runtime.h>
#include <hip/hip_bf16.h>

// Problem sizes (fixed by the reference): x[B,N,C], proj_w[CR,C], out[B,1,CR]
#define B_   8
#define N_   4096
#define C_   256
#define CR_  64
#define LN_EPS 1e-5f

typedef float v2f __attribute__((ext_vector_type(2)));
typedef float v8f __attribute__((ext_vector_type(8)));

// ---------------- wave32 reductions ----------------
__device__ __forceinline__ float wave_sum(float v) {
#pragma unroll
    for (int off = 16; off >= 1; off >>= 1) v += __shfl_xor(v, off, 32);
    return v;
}
__device__ __forceinline__ float wave_max(float v) {
#pragma unroll
    for (int off = 16; off >= 1; off >>= 1) v = fmaxf(v, __shfl_xor(v, off, 32));
    return v;
}

// One wave handles one row of C=256 floats: lane owns channels [lane*8, lane*8+8)
__device__ __forceinline__ void ln_row(const float* __restrict__ xrow,
                                       const float* wv, const float* bv,
                                       int lane, float* y) {
    float v[8];
    const float4* p = reinterpret_cast<const float4*>(xrow) + lane * 2;
    float4 q0 = p[0], q1 = p[1];
    v[0]=q0.x; v[1]=q0.y; v[2]=q0.z; v[3]=q0.w;
    v[4]=q1.x; v[5]=q1.y; v[6]=q1.z; v[7]=q1.w;
    float s1 = 0.f, s2 = 0.f;
#pragma unroll
    for (int j = 0; j < 8; ++j) { s1 += v[j]; s2 += v[j]*v[j]; }
    s1 = wave_sum(s1);
    s2 = wave_sum(s2);
    float mu   = s1 * (1.f / C_);
    float var  = s2 * (1.f / C_) - mu * mu;
    float rstd = rsqrtf(var + LN_EPS);
#pragma unroll
    for (int j = 0; j < 8; ++j) y[j] = (v[j] - mu) * rstd * wv[j] + bv[j];
}

// ---------------- kernel 0: zero scratch that is accumulated into ----------------
__global__ void k_init(float* __restrict__ center) {
    for (int i = threadIdx.x; i < 16 * C_; i += blockDim.x) center[i] = 0.f;
}

// ---------------- kernel 1: LN stats: rinv[b,n] and per-block partial S ----------------
// grid(32, B) x 256 threads; block handles 128 rows (8 waves x 16 rows)
__global__ void k_ln_stats(const float* __restrict__ x,
                           const float* __restrict__ lnw,
                           const float* __restrict__ lnb,
                           float* __restrict__ rinv,
                           float* __restrict__ Spart) {
    const int b = blockIdx.y, chunk = blockIdx.x;
    const int lane = threadIdx.x & 31, w = threadIdx.x >> 5;
    float wv[8], bv[8];
#pragma unroll
    for (int j = 0; j < 8; ++j) { wv[j] = lnw[lane*8 + j]; bv[j] = lnb[lane*8 + j]; }
    float acc[8];
#pragma unroll
    for (int j = 0; j < 8; ++j) acc[j] = 0.f;

    for (int i = 0; i < 16; ++i) {
        const int n = chunk * 128 + i * 8 + w;
        const float* xrow = x + ((size_t)b * N_ + n) * C_;
        float y[8];
        ln_row(xrow, wv, bv, lane, y);
        float r2 = 0.f;
#pragma unroll
        for (int j = 0; j < 8; ++j) r2 += y[j] * y[j];
        r2 = wave_sum(r2);
        const float ri = rsqrtf(r2);          // 1/||y_n|| ; eps in cos-sim denom is ~4e-11 rel, dropped
        if (lane == 0) rinv[b * N_ + n] = ri;
#pragma unroll
        for (int j = 0; j < 8; ++j) acc[j] += y[j] * ri;
    }
    __shared__ float Sloc[8][C_];
#pragma unroll
    for (int j = 0; j < 8; ++j) Sloc[w][lane*8 + j] = acc[j];
    __syncthreads();
    const int tid = threadIdx.x;
    float t = 0.f;
#pragma unroll
    for (int ww = 0; ww < 8; ++ww) t += Sloc[ww][tid];
    Spart[((size_t)b * 32 + chunk) * C_ + tid] = t;   // deterministic (no float atomics)
}

// ---------------- kernel 2: reduce partial S -> S[B,C] ----------------
__global__ void k_reduce_S(const float* __restrict__ Spart, float* __restrict__ S) {
    const int b = blockIdx.x, c = threadIdx.x;
    float t = 0.f;
    for (int chunk = 0; chunk < 32; ++chunk)
        t += Spart[((size_t)b * 32 + chunk) * C_ + c];
    S[b * C_ + c] = t;
}

// ---------------- kernel 3: density[b,n] = (y_n . S_b) * rinv[b,n] ----------------
__global__ void k_density(const float* __restrict__ x,
                          const float* __restrict__ lnw,
                          const float* __restrict__ lnb,
                          const float* __restrict__ rinv,
                          const float* __restrict__ S,
                          float* __restrict__ density) {
    const int b = blockIdx.y, chunk = blockIdx.x;
    const int lane = threadIdx.x & 31, w = threadIdx.x >> 5;
    float wv[8], bv[8], Sl[8];
#pragma unroll
    for (int j = 0; j < 8; ++j) {
        wv[j] = lnw[lane*8 + j]; bv[j] = lnb[lane*8 + j];
        Sl[j] = S[b * C_ + lane*8 + j];
    }
    for (int i = 0; i < 16; ++i) {
        const int n = chunk * 128 + i * 8 + w;
        const float* xrow = x + ((size_t)b * N_ + n) * C_;
        float y[8];
        ln_row(xrow, wv, bv, lane, y);
        float d = 0.f;
#pragma unroll
        for (int j = 0; j < 8; ++j) d += y[j] * Sl[j];
        d = wave_sum(d);
        if (lane == 0) density[b * N_ + n] = d * rinv[b * N_ + n];
    }
}

// ---------------- kernel 4: per-batch max of density ----------------
__global__ void k_max(const float* __restrict__ density, float* __restrict__ maxd) {
    const int b = blockIdx.x, tid = threadIdx.x;
    const int lane = tid & 31, w = tid >> 5;
    float m = -__builtin_inff();
    for (int i = 0; i < N_ / 256; ++i)
        m = fmaxf(m, density[b * N_ + i * 256 + tid]);
    m = wave_max(m);
    __shared__ float red[8];
    if (lane == 0) red[w] = m;
    __syncthreads();
    if (tid == 0) {
        float mm = red[0];
#pragma unroll
        for (int ww = 1; ww < 8; ++ww) mm = fmaxf(mm, red[ww]);
        maxd[b] = mm;
    }
}

// ---------------- kernel 5: sum LN rows where density == max (tie-aware) ----------------
__global__ void k_select(const float* __restrict__ x,
                         const float* __restrict__ lnw,
                         const float* __restrict__ lnb,
                         const float* __restrict__ density,
                         const float* __restrict__ maxd,
                         float* __restrict__ center) {
    const int b = blockIdx.y, chunk = blockIdx.x;
    const int lane = threadIdx.x & 31, w = threadIdx.x >> 5;
    const float md = maxd[b];
    float wv[8], bv[8];
#pragma unroll
    for (int j = 0; j < 8; ++j) { wv[j] = lnw[lane*8 + j]; bv[j] = lnb[lane*8 + j]; }
    for (int i = 0; i < 16; ++i) {
        const int n = chunk * 128 + i * 8 + w;
        if (density[b * N_ + n] == md) {           // wave-uniform branch
            const float* xrow = x + ((size_t)b * N_ + n) * C_;
            float y[8];
            ln_row(xrow, wv, bv, lane, y);
#pragma unroll
            for (int j = 0; j < 8; ++j)
                atomicAdd(&center[b * C_ + lane*8 + j], y[j]);  // ~1 row/batch
        }
    }
}

// ---------------- kernel 6: out = relu(center @ W^T + bias) via V_WMMA_F32_16X16X4_F32 ----
// center: [16][C] fp32 (rows 8..15 zero-padded), W: [CR][C], out: [B][CR]
// 1 block x 128 threads = 4 waves; wave w owns output columns [16w, 16w+16)
__global__ void k_proj(const float* __restrict__ center,
                       const float* __restrict__ W,
                       const float* __restrict__ bias,
                       float* __restrict__ out) {
    const int lane = threadIdx.x & 31;
    const int wv   = threadIdx.x >> 5;
    const int half = lane >> 4;          // A/B fragment lane half (K split)
    const int l16  = lane & 15;
    const int n    = wv * 16 + l16;      // output column 0..63

    v8f acc = {};
    for (int k0 = 0; k0 < C_; k0 += 4) {
        const int ka = k0 + 2 * half;
        v2f a, bf;
        // A 16x4 (MxK): lanes 0-15 hold K0,K1; lanes 16-31 hold K2,K3 (M = lane&15)
        a.x = center[l16 * C_ + ka];
        a.y = center[l16 * C_ + ka + 1];
        // B 4x16 (KxN): B[k][n] = W[n][k]; same K-half split across lane halves
        bf.x = W[n * C_ + ka];
        bf.y = W[n * C_ + ka + 1];
        acc = __builtin_amdgcn_wmma_f32_16x16x4_f32(
            /*neg_a=*/false, a, /*neg_b=*/false, bf,
            /*c_mod=*/(short)0, acc, /*reuse_a=*/false, /*reuse_b=*/false);
    }
    // D 16x16 f32: VGPR v -> M = v (lanes 0-15) or v+8 (lanes 16-31), N = lane&15 tile col
#pragma unroll
    for (int v = 0; v < 8; ++v) {
        const int m = v + 8 * half;
        if (m < B_) {
            float r = acc[v] + bias[n];
            out[m * CR_ + n] = r > 0.f ? r : 0.f;
        }
    }
}

extern "C" void kernel_launch(void* const* d_in, const int* in_sizes, int n_in,
                              void* d_out, int out_size, void* d_ws, size_t ws_size,
                              hipStream_t stream) {
    (void)in_sizes; (void)n_in; (void)out_size; (void)ws_size;
    const float* x     = (const float*)d_in[0];
    const float* ln_w  = (const float*)d_in[1];
    const float* ln_b  = (const float*)d_in[2];
    const float* projw = (const float*)d_in[3];
    const float* projb = (const float*)d_in[4];
    float* out = (float*)d_out;

    // workspace layout (floats): ~549 KB total
    float* ws      = (float*)d_ws;
    float* rinv    = ws;                          // B*N
    float* density = rinv    + (size_t)B_ * N_;   // B*N
    float* Spart   = density + (size_t)B_ * N_;   // B*32*C
    float* S       = Spart   + (size_t)B_ * 32 * C_; // B*C
    float* center  = S       + (size_t)B_ * C_;   // 16*C (padded for WMMA M=16)
    float* maxd    = center  + (size_t)16 * C_;   // B

    dim3 rowsGrid(32, B_);
    k_init    <<<1,        256, 0, stream>>>(center);
    k_ln_stats<<<rowsGrid, 256, 0, stream>>>(x, ln_w, ln_b, rinv, Spart);
    k_reduce_S<<<B_,       256, 0, stream>>>(Spart, S);
    k_density <<<rowsGrid, 256, 0, stream>>>(x, ln_w, ln_b, rinv, S, density);
    k_max     <<<B_,       256, 0, stream>>>(density, maxd);
    k_select  <<<rowsGrid, 256, 0, stream>>>(x, ln_w, ln_b, density, maxd, center);
    k_proj    <<<1,        128, 0, stream>>>(center, projw, projb, out);
}